// HyperConnectionsCUDA_44667659879107
// MI455X (gfx1250) — compile-verified
//
#include <hip/hip_runtime.h>
#include <math.h>

// ---------------------------------------------------------------------------
// Problem constants (from reference)
// ---------------------------------------------------------------------------
#define S_DIM 4
#define B_DIM 4
#define T_DIM 4096
#define D_DIM 1024
#define TD    (T_DIM * D_DIM)      // stride between streams: 4,194,304 elems
#define STD   (S_DIM * TD)         // stride between batches
#define NTOK  (B_DIM * T_DIM)      // 16384 tokens
#define SK_ITERS 20
#define EPS 1e-5f

#define BLK   256                  // 8 wave32s
#define NWAVE (BLK / 32)
#define GRID  2048                 // 8 tokens per block -> pipeline has depth

typedef float v4f __attribute__((ext_vector_type(4)));
typedef int   v4i __attribute__((vector_size(16)));   // matches builtin pointee

#define AS1 __attribute__((address_space(1)))
#define AS3 __attribute__((address_space(3)))

// ---- CDNA5 async-copy path (guarded; falls back to direct loads) ----------
#if defined(__has_builtin)
# if __has_builtin(__builtin_amdgcn_global_load_async_to_lds_b128)
#  define HAVE_ASYNC_LDS 1
# endif
#endif
#ifndef HAVE_ASYNC_LDS
# define HAVE_ASYNC_LDS 0
#endif

#if defined(__has_builtin)
# if __has_builtin(__builtin_amdgcn_s_wait_asynccnt)
#  define WAIT_ASYNCCNT(n) __builtin_amdgcn_s_wait_asynccnt(n)
# else
#  define WAIT_ASYNCCNT(n) asm volatile("s_wait_asynccnt %0" ::"i"(n) : "memory")
# endif
#else
# define WAIT_ASYNCCNT(n) asm volatile("s_wait_asynccnt %0" ::"i"(n) : "memory")
#endif

#if HAVE_ASYNC_LDS
__device__ __forceinline__ void async_cp16(const float* gsrc, void* lds) {
  // global -> LDS, 16B per lane, tracked on ASYNCcnt
  __builtin_amdgcn_global_load_async_to_lds_b128(
      (AS1 v4i*)(unsigned long long)(const void*)gsrc,
      (AS3 v4i*)(unsigned int)(unsigned long long)lds,
      /*offset=*/0, /*cpol=*/0);
}
#endif

// ---------------------------------------------------------------------------
// Per-token compute: aggregate, RMS-reduce across block, write 4 streams
// ---------------------------------------------------------------------------
__device__ __forceinline__ void token_compute_store(
    const v4f* a, const float* hpre, const float* cmix, v4f w4,
    float* red, int lane, int wid, float* __restrict__ obase) {
  v4f agg = hpre[0] * a[0] + hpre[1] * a[1] + hpre[2] * a[2] + hpre[3] * a[3];
  float ssq = agg.x * agg.x + agg.y * agg.y + agg.z * agg.z + agg.w * agg.w;
#pragma unroll
  for (int off = 16; off > 0; off >>= 1)
    ssq += __shfl_xor(ssq, off, 32);
  if (lane == 0) red[wid] = ssq;
  __syncthreads();
  float tot = 0.0f;
#pragma unroll
  for (int wv = 0; wv < NWAVE; ++wv) tot += red[wv];
  float inv_rms = 1.0f / sqrtf(tot * (1.0f / (float)D_DIM) + EPS);
  v4f xn = (agg * inv_rms) * w4;
#pragma unroll
  for (int s = 0; s < S_DIM; ++s)
    *(v4f*)(obase + (size_t)s * TD) = a[s] + cmix[s] * xn;
  __syncthreads();  // red[] is reused by the next token
}

// ---------------------------------------------------------------------------
// Main kernel (defined FIRST so the disasm snippet shows it). Streaming,
// grid-stride over tokens; each thread owns a float4 of D. Async
// double-buffered global->LDS loads (ASYNCcnt pipelining); per-lane-private
// LDS regions so only per-wave s_wait_asynccnt is needed for the tile.
// ---------------------------------------------------------------------------
__global__ __launch_bounds__(BLK) void hc_main_kernel(
    const float* __restrict__ resid,
    const float* __restrict__ weight,
    const float* __restrict__ params,
    float* __restrict__ out) {
  const int tid  = threadIdx.x;
  const int lane = tid & 31;
  const int wid  = tid >> 5;
  const int d0   = tid << 2;

  __shared__ float red[NWAVE];
#if HAVE_ASYNC_LDS
  __shared__ v4f xbuf[2][S_DIM][BLK];  // 32 KB double buffer
#endif

  v4f w4 = *(const v4f*)(weight + d0);
  float hpre[S_DIM], cmix[S_DIM];
#pragma unroll
  for (int s = 0; s < S_DIM; ++s) {
    hpre[s] = params[s];
    cmix[s] = params[S_DIM + s];
  }

#if HAVE_ASYNC_LDS
  {  // prologue: issue token blockIdx.x into buffer 0
    const int n0 = blockIdx.x;
    const float* base =
        resid + (size_t)(n0 >> 12) * STD + (size_t)(n0 & (T_DIM - 1)) * D_DIM + d0;
#pragma unroll
    for (int s = 0; s < S_DIM; ++s)
      async_cp16(base + (size_t)s * TD, (void*)&xbuf[0][s][tid]);
  }
  int buf = 0;
  for (int n = blockIdx.x; n < NTOK; n += GRID) {
    const int nn = n + GRID;
    if (nn < NTOK) {
      const float* base =
          resid + (size_t)(nn >> 12) * STD + (size_t)(nn & (T_DIM - 1)) * D_DIM + d0;
#pragma unroll
      for (int s = 0; s < S_DIM; ++s)
        async_cp16(base + (size_t)s * TD, (void*)&xbuf[buf ^ 1][s][tid]);
      WAIT_ASYNCCNT(4);  // 4 newest still in flight; current token's 4 done
    } else {
      WAIT_ASYNCCNT(0);
    }
    v4f a[S_DIM];
#pragma unroll
    for (int s = 0; s < S_DIM; ++s) a[s] = xbuf[buf][s][tid];
    float* obase =
        out + (size_t)(n >> 12) * STD + (size_t)(n & (T_DIM - 1)) * D_DIM + d0;
    token_compute_store(a, hpre, cmix, w4, red, lane, wid, obase);
    buf ^= 1;
  }
#else
  for (int n = blockIdx.x; n < NTOK; n += GRID) {
    const float* base =
        resid + (size_t)(n >> 12) * STD + (size_t)(n & (T_DIM - 1)) * D_DIM + d0;
    v4f a[S_DIM];
#pragma unroll
    for (int s = 0; s < S_DIM; ++s)
      a[s] = *(const v4f*)(base + (size_t)s * TD);
    const int nn = n + GRID;
    if (nn < NTOK) {  // prefetch next iteration's lines (global_prefetch_b8)
      const float* nbase =
          resid + (size_t)(nn >> 12) * STD + (size_t)(nn & (T_DIM - 1)) * D_DIM + d0;
#pragma unroll
      for (int s = 0; s < S_DIM; ++s)
        __builtin_prefetch((const void*)(nbase + (size_t)s * TD), 0, 1);
    }
    float* obase =
        out + (size_t)(n >> 12) * STD + (size_t)(n & (T_DIM - 1)) * D_DIM + d0;
    token_compute_store(a, hpre, cmix, w4, red, lane, wid, obase);
  }
#endif
}

// ---------------------------------------------------------------------------
// Prep kernel: Sinkhorn on 4x4 + gate precompute (token-independent scalars)
//   params[0..3] = h_pre[s] = sigmoid(H_pre_logits[s])
//   params[4..7] = c[s]     = sum_r M[s][r] * 2*sigmoid(H_post_logits[r])
// Runs once on a single lane; keep loops rolled to minimize code bloat.
// ---------------------------------------------------------------------------
__global__ void hc_prep_kernel(const float* __restrict__ hpre_l,
                               const float* __restrict__ hpost_l,
                               const float* __restrict__ hres,
                               float* __restrict__ params) {
  if (threadIdx.x != 0) return;
  float M[S_DIM][S_DIM];
  for (int i = 0; i < S_DIM; ++i)
    for (int j = 0; j < S_DIM; ++j)
      M[i][j] = expf(hres[i * S_DIM + j]);
#pragma clang loop unroll(disable)
  for (int it = 0; it < SK_ITERS; ++it) {
    for (int i = 0; i < S_DIM; ++i) {
      float r = M[i][0] + M[i][1] + M[i][2] + M[i][3] + EPS;
      float rr = 1.0f / r;
      for (int j = 0; j < S_DIM; ++j) M[i][j] *= rr;
    }
    for (int j = 0; j < S_DIM; ++j) {
      float cs = M[0][j] + M[1][j] + M[2][j] + M[3][j] + EPS;
      float rc = 1.0f / cs;
      for (int i = 0; i < S_DIM; ++i) M[i][j] *= rc;
    }
  }
  float hpost[S_DIM];
  for (int s = 0; s < S_DIM; ++s)
    hpost[s] = 2.0f / (1.0f + expf(-hpost_l[s]));
  for (int s = 0; s < S_DIM; ++s) {
    params[s] = 1.0f / (1.0f + expf(-hpre_l[s]));
    float c = 0.0f;
    for (int r = 0; r < S_DIM; ++r) c += M[s][r] * hpost[r];
    params[S_DIM + s] = c;
  }
}

// ---------------------------------------------------------------------------
extern "C" void kernel_launch(void* const* d_in, const int* in_sizes, int n_in,
                              void* d_out, int out_size, void* d_ws, size_t ws_size,
                              hipStream_t stream) {
  const float* resid  = (const float*)d_in[0];  // residuals   (B*S, T, D)
  const float* weight = (const float*)d_in[1];  // rmsnorm_weight (D)
  const float* hpre_l = (const float*)d_in[2];  // H_pre_logits (S)
  const float* hpost_l= (const float*)d_in[3];  // H_post_logits (S)
  const float* hres   = (const float*)d_in[4];  // H_res (S,S)
  float* params = (float*)d_ws;                 // 8 floats scratch
  float* outp   = (float*)d_out;

  hipLaunchKernelGGL(hc_prep_kernel, dim3(1), dim3(32), 0, stream,
                     hpre_l, hpost_l, hres, params);
  hipLaunchKernelGGL(hc_main_kernel, dim3(GRID), dim3(BLK), 0, stream,
                     resid, weight, params, outp);
}